// SEBlock_60610578481199
// MI455X (gfx1250) — compile-verified
//
#include <hip/hip_runtime.h>

typedef float v2f __attribute__((ext_vector_type(2)));
typedef float v4f __attribute__((ext_vector_type(4)));
typedef float v8f __attribute__((ext_vector_type(8)));

#define BB  32
#define CC  256
#define HID 16
#define HW  4096  // 64*64

// ---------------- Phase 1: squeeze — s[b,c] = mean over H*W ----------------
// One 256-thread block per (b,c); each channel is a contiguous 16 KiB run.
__global__ void __launch_bounds__(256) se_squeeze(const float* __restrict__ x,
                                                  float* __restrict__ s) {
  const int bc = blockIdx.x;  // 0..8191
  const v4f* p4 = (const v4f*)(x + (size_t)bc * HW);
  float sum = 0.0f;
  for (int i = threadIdx.x; i < HW / 4; i += 256) {
    v4f v = p4[i];
    sum += (v.x + v.y) + (v.z + v.w);
  }
  __shared__ float red[256];
  red[threadIdx.x] = sum;
  __syncthreads();
  for (int off = 128; off > 0; off >>= 1) {
    if (threadIdx.x < off) red[threadIdx.x] += red[threadIdx.x + off];
    __syncthreads();
  }
  if (threadIdx.x == 0) s[bc] = red[0] * (1.0f / (float)HW);
}

// ---------------- Phase 2: excite MLP via V_WMMA_F32_16X16X4_F32 ----------------
// h = relu(s @ w1^T + b1)  [32,16];  g = sigmoid(h @ w2^T + b2)  [32,256]
// One block, 64 threads (2 full wave32s); wave t owns rows 16t..16t+15.
__global__ void __launch_bounds__(64) se_mlp(const float* __restrict__ s,
                                             const float* __restrict__ w1,
                                             const float* __restrict__ b1,
                                             const float* __restrict__ w2,
                                             const float* __restrict__ b2,
                                             float* __restrict__ g) {
  __shared__ float hS[BB][HID + 1];  // +1 pad vs LDS bank conflicts
  const int lane = threadIdx.x & 31;
  const int t    = threadIdx.x >> 5;  // m-tile (wave id): 0 or 1
  const int l16  = lane & 15;         // M (for A) / N (for B,C,D) within tile
  const int kg   = lane >> 4;         // K-group: lanes 16..31 carry K+2,K+3
  const int row  = t * 16 + l16;

  // ---- Step A: h-pre = s[32,256] x w1^T[256,16], K in chunks of 4 ----
  v8f c = {};
  for (int k0 = 0; k0 < CC; k0 += 4) {
    const int k = k0 + 2 * kg;
    v2f a, b;
    // A 16x4 f32 layout: lane l16 = M; VGPR j holds K = k0 + 2*kg + j
    a.x = s[row * CC + k];
    a.y = s[row * CC + k + 1];
    // B 4x16: lane l16 = N; B[k][n] = w1[n][k]  (w1 is [HID, C] row-major)
    b.x = w1[l16 * CC + k];
    b.y = w1[l16 * CC + k + 1];
    c = __builtin_amdgcn_wmma_f32_16x16x4_f32(false, a, false, b,
                                              (short)0, c, false, false);
  }
  // C/D 16x16 f32: VGPR r => M = r + 8*kg, N = l16
#pragma unroll
  for (int r = 0; r < 8; ++r) {
    const int m = t * 16 + r + 8 * kg;
    hS[m][l16] = fmaxf(c[r] + b1[l16], 0.0f);
  }
  __syncthreads();

  // ---- Step B: g-pre = h[32,16] x w2^T[16,256], 16 N-tiles ----
  for (int nt = 0; nt < CC / 16; ++nt) {
    const int n = nt * 16 + l16;
    v8f c2 = {};
#pragma unroll
    for (int k0 = 0; k0 < HID; k0 += 4) {
      const int k = k0 + 2 * kg;
      v2f a, b;
      a.x = hS[row][k];
      a.y = hS[row][k + 1];
      // B[k][n] = w2[n][k]  (w2 is [C, HID] row-major)
      b.x = w2[n * HID + k];
      b.y = w2[n * HID + k + 1];
      c2 = __builtin_amdgcn_wmma_f32_16x16x4_f32(false, a, false, b,
                                                 (short)0, c2, false, false);
    }
#pragma unroll
    for (int r = 0; r < 8; ++r) {
      const int m = t * 16 + r + 8 * kg;
      const float v = c2[r] + b2[n];
      g[m * CC + n] = 1.0f / (1.0f + __expf(-v));
    }
  }
}

// ---------------- Phase 3: scale — out = x * g[b,c] ----------------
// x re-read should hit the 192 MB L2 (x is 128 MiB). Output stored
// non-temporally so it does not evict x's L2 lines mid-pass.
__global__ void __launch_bounds__(256) se_scale(const float* __restrict__ x,
                                                const float* __restrict__ g,
                                                float* __restrict__ out) {
  const size_t i4 = (size_t)blockIdx.x * 256 + threadIdx.x;
  const int bc = (int)(i4 >> 10);  // 1024 float4s per (b,c) channel
  const float gv = g[bc];          // uniform within a block (1024 % 256 == 0)
  v4f v = ((const v4f*)x)[i4];
  v.x *= gv; v.y *= gv; v.z *= gv; v.w *= gv;
  __builtin_nontemporal_store(v, ((v4f*)out) + i4);
}

extern "C" void kernel_launch(void* const* d_in, const int* in_sizes, int n_in,
                              void* d_out, int out_size, void* d_ws, size_t ws_size,
                              hipStream_t stream) {
  const float* x  = (const float*)d_in[0];   // [32,256,64,64]
  const float* w1 = (const float*)d_in[1];   // [16,256]
  const float* b1 = (const float*)d_in[2];   // [16]
  const float* w2 = (const float*)d_in[3];   // [256,16]
  const float* b2 = (const float*)d_in[4];   // [256]
  float* out = (float*)d_out;

  float* s = (float*)d_ws;      // 8192 floats
  float* g = s + BB * CC;       // 8192 floats

  se_squeeze<<<BB * CC, 256, 0, stream>>>(x, s);
  se_mlp<<<1, 64, 0, stream>>>(s, w1, b1, w2, b2, g);
  const int n4 = BB * CC * HW / 4;  // 8,388,608 float4s
  se_scale<<<n4 / 256, 256, 0, stream>>>(x, g, out);
}